// Lut3D_52810917872156
// MI455X (gfx1250) — compile-verified
//
#include <hip/hip_runtime.h>
#include <hip/hip_bf16.h>
#include <stdint.h>

#define LUT_DIM   33
#define LUT_N     (LUT_DIM*LUT_DIM*LUT_DIM)   /* 35937 */
#define IMG_S     (1920*1080)                 /* 2073600 */
#define N_IMG     8
#define N_PIX     (N_IMG*IMG_S)               /* 16588800, divisible by 32 */

#ifndef __has_builtin
#define __has_builtin(x) 0
#endif

#if __has_builtin(__builtin_amdgcn_global_load_async_to_lds_b32) && \
    __has_builtin(__builtin_amdgcn_s_wait_asynccnt)
#define HAVE_ASYNC_LDS 1
#else
#define HAVE_ASYNC_LDS 0
#endif

#if HAVE_ASYNC_LDS
typedef __attribute__((address_space(1))) int* as1_int_p;
typedef __attribute__((address_space(3))) int* as3_int_p;
#endif

// ---------------------------------------------------------------------------
// Helpers
// ---------------------------------------------------------------------------
__device__ __forceinline__ void lut_decode(float v, int& idx, float& fr) {
    // reference: binsize = 1.000001/32 ; xb = v / binsize
    const float inv_bin = 32.0f / 1.000001f;
    float xb = v * inv_bin;
    int i = (int)floorf(xb);
    i = i < 0 ? 0 : i;
    i = i > (LUT_DIM - 2) ? (LUT_DIM - 2) : i;
    idx = i;
    fr  = xb - (float)i;
}

__device__ __forceinline__ float4 lerp4(float4 a, float4 b, float t) {
    float4 r;
    r.x = fmaf(t, b.x - a.x, a.x);
    r.y = fmaf(t, b.y - a.y, a.y);
    r.z = fmaf(t, b.z - a.z, a.z);
    r.w = 0.0f;
    return r;
}

__device__ __forceinline__ void compute_store(const float4* __restrict__ lutp,
                                              float* __restrict__ out,
                                              int p, float r, float g, float b)
{
    int ri, gi, bi; float rd, gd, bd;
    lut_decode(r, ri, rd);
    lut_decode(g, gi, gd);
    lut_decode(b, bi, bd);
    const int base = (bi * LUT_DIM + gi) * LUT_DIM + ri;

    // 8 corner fetches, each one B128 from the L2-resident packed LUT
    const float4 c000 = lutp[base];
    const float4 c100 = lutp[base + 1];
    const float4 c010 = lutp[base + LUT_DIM];
    const float4 c110 = lutp[base + LUT_DIM + 1];
    const float4 c001 = lutp[base + LUT_DIM * LUT_DIM];
    const float4 c101 = lutp[base + LUT_DIM * LUT_DIM + 1];
    const float4 c011 = lutp[base + LUT_DIM * LUT_DIM + LUT_DIM];
    const float4 c111 = lutp[base + LUT_DIM * LUT_DIM + LUT_DIM + 1];

    float4 c00 = lerp4(c000, c100, rd);
    float4 c10 = lerp4(c010, c110, rd);
    float4 c01 = lerp4(c001, c101, rd);
    float4 c11 = lerp4(c011, c111, rd);
    float4 c0  = lerp4(c00, c10, gd);
    float4 c1  = lerp4(c01, c11, gd);
    float4 cc  = lerp4(c0, c1, bd);

    const int n = p / IMG_S;
    const int i = p - n * IMG_S;
    const size_t o = (size_t)n * IMG_S + i;
    // Non-temporal stores: don't let the 199MB output stream evict the LUT
    // from L2 (store TH=NT policy on gfx1250).
    __builtin_nontemporal_store(cc.x, &out[o]);
    __builtin_nontemporal_store(cc.y, &out[(size_t)N_IMG * IMG_S + o]);
    __builtin_nontemporal_store(cc.z, &out[2 * (size_t)N_IMG * IMG_S + o]);
}

// ---------------------------------------------------------------------------
// Kernel 1: repack channel-planar LUT -> interleaved padded float4 texels
// ---------------------------------------------------------------------------
__global__ __launch_bounds__(256)
void lut_pack_kernel(const float* __restrict__ lut, float4* __restrict__ lutp)
{
    int t = blockIdx.x * 256 + threadIdx.x;
    if (t < LUT_N) {
        float4 v;
        v.x = lut[t];
        v.y = lut[LUT_N + t];
        v.z = lut[2 * LUT_N + t];
        v.w = 0.0f;
        lutp[t] = v;
    }
}

// ---------------------------------------------------------------------------
// Kernel 2: main apply kernel. Per-wave double-buffered async staging of the
// input stream through LDS (ASYNCcnt pipeline), gathers from packed LUT.
// ---------------------------------------------------------------------------
#if HAVE_ASYNC_LDS
__device__ __forceinline__ void stage_chunk(const float* __restrict__ x,
                                            float* wbuf, int slot, int k, int lane)
{
    const int p = k * 32 + lane;
    const int n = p / IMG_S;
    const int i = p - n * IMG_S;
    const float* src = x + (size_t)n * (3 * IMG_S) + i;
    float* dst = wbuf + slot * 96 + lane;
    __builtin_amdgcn_global_load_async_to_lds_b32(
        (as1_int_p)(src), (as3_int_p)(dst), 0, 0);
    __builtin_amdgcn_global_load_async_to_lds_b32(
        (as1_int_p)(src + IMG_S), (as3_int_p)(dst + 32), 0, 0);
    __builtin_amdgcn_global_load_async_to_lds_b32(
        (as1_int_p)(src + 2 * IMG_S), (as3_int_p)(dst + 64), 0, 0);
}
#endif

__global__ __launch_bounds__(256)
void lut3d_apply_packed(const float4* __restrict__ lutp,
                        const float*  __restrict__ x,
                        float*        __restrict__ out,
                        int nChunks)
{
#if HAVE_ASYNC_LDS
    __shared__ float stage[8 * 2 * 96];   // 8 waves x 2 slots x (3ch x 32 lanes)
#endif
    const int lane       = threadIdx.x & 31;
    const int wid        = threadIdx.x >> 5;
    const int wavesTotal = gridDim.x * 8;
    const int gwave      = blockIdx.x * 8 + wid;

    // Warm L2 with the packed LUT: one global_prefetch_b8 per 64B line,
    // issued once across the grid (first ~36 blocks only).
    {
        const size_t off = (size_t)(blockIdx.x * 256 + threadIdx.x) * 64;
        if (off < (size_t)LUT_N * sizeof(float4))
            __builtin_prefetch((const char*)lutp + off, 0, 0);
    }

#if HAVE_ASYNC_LDS
    float* wbuf = &stage[wid * 192];
    int k = gwave;
    int slot = 0;
    if (k < nChunks)
        stage_chunk(x, wbuf, 0, k, lane);
    for (; k < nChunks; k += wavesTotal) {
        const int kn = k + wavesTotal;
        if (kn < nChunks) {
            stage_chunk(x, wbuf, slot ^ 1, kn, lane);   // prefetch next chunk
            __builtin_amdgcn_s_wait_asynccnt(3);        // current chunk's 3 done
        } else {
            __builtin_amdgcn_s_wait_asynccnt(0);
        }
        asm volatile("" ::: "memory");
        const float r = wbuf[slot * 96 + 0  + lane];
        const float g = wbuf[slot * 96 + 32 + lane];
        const float b = wbuf[slot * 96 + 64 + lane];
        compute_store(lutp, out, k * 32 + lane, r, g, b);
        slot ^= 1;
    }
#else
    for (int k = gwave; k < nChunks; k += wavesTotal) {
        const int p = k * 32 + lane;
        const int n = p / IMG_S;
        const int i = p - n * IMG_S;
        const float* xs = x + (size_t)n * (3 * IMG_S) + i;
        const float r = __builtin_nontemporal_load(&xs[0]);
        const float g = __builtin_nontemporal_load(&xs[IMG_S]);
        const float b = __builtin_nontemporal_load(&xs[2 * IMG_S]);
        compute_store(lutp, out, p, r, g, b);
    }
#endif
}

// ---------------------------------------------------------------------------
// Fallback kernel when workspace is too small for the packed LUT:
// gather straight from the channel-planar LUT (24 scalar gathers/pixel).
// ---------------------------------------------------------------------------
__device__ __forceinline__ float lerp1(float a, float b, float t) {
    return fmaf(t, b - a, a);
}

__global__ __launch_bounds__(256)
void lut3d_apply_direct(const float* __restrict__ lut,
                        const float* __restrict__ x,
                        float*       __restrict__ out)
{
    const int p = blockIdx.x * 256 + threadIdx.x;
    if (p >= N_PIX) return;
    const int n = p / IMG_S;
    const int i = p - n * IMG_S;
    const float* xs = x + (size_t)n * (3 * IMG_S) + i;
    const float r = __builtin_nontemporal_load(&xs[0]);
    const float g = __builtin_nontemporal_load(&xs[IMG_S]);
    const float b = __builtin_nontemporal_load(&xs[2 * IMG_S]);

    int ri, gi, bi; float rd, gd, bd;
    lut_decode(r, ri, rd);
    lut_decode(g, gi, gd);
    lut_decode(b, bi, bd);
    const int base = (bi * LUT_DIM + gi) * LUT_DIM + ri;
    const size_t o = (size_t)n * IMG_S + i;

    #pragma unroll
    for (int c = 0; c < 3; ++c) {
        const float* L = lut + (size_t)c * LUT_N;
        const float v000 = L[base];
        const float v100 = L[base + 1];
        const float v010 = L[base + LUT_DIM];
        const float v110 = L[base + LUT_DIM + 1];
        const float v001 = L[base + LUT_DIM * LUT_DIM];
        const float v101 = L[base + LUT_DIM * LUT_DIM + 1];
        const float v011 = L[base + LUT_DIM * LUT_DIM + LUT_DIM];
        const float v111 = L[base + LUT_DIM * LUT_DIM + LUT_DIM + 1];
        const float v00 = lerp1(v000, v100, rd);
        const float v10 = lerp1(v010, v110, rd);
        const float v01 = lerp1(v001, v101, rd);
        const float v11 = lerp1(v011, v111, rd);
        const float v0  = lerp1(v00, v10, gd);
        const float v1  = lerp1(v01, v11, gd);
        __builtin_nontemporal_store(lerp1(v0, v1, bd),
                                    &out[(size_t)c * N_IMG * IMG_S + o]);
    }
}

// ---------------------------------------------------------------------------
// Host launcher
// ---------------------------------------------------------------------------
extern "C" void kernel_launch(void* const* d_in, const int* in_sizes, int n_in,
                              void* d_out, int out_size, void* d_ws, size_t ws_size,
                              hipStream_t stream)
{
    const float* lut = (const float*)d_in[0];   // (3, 33, 33, 33) fp32
    const float* x   = (const float*)d_in[1];   // (8, 3, 1920, 1080) fp32
    float* out = (float*)d_out;                 // (3, 8, 1920, 1080) fp32

    if (ws_size >= (size_t)LUT_N * sizeof(float4)) {
        float4* lutp = (float4*)d_ws;
        lut_pack_kernel<<<(LUT_N + 255) / 256, 256, 0, stream>>>(lut, lutp);
        const int nChunks = N_PIX / 32;         // 518400
        lut3d_apply_packed<<<2048, 256, 0, stream>>>(lutp, x, out, nChunks);
    } else {
        lut3d_apply_direct<<<(N_PIX + 255) / 256, 256, 0, stream>>>(lut, x, out);
    }
}